// BaseModel_27891517620526
// MI455X (gfx1250) — compile-verified
//
#include <hip/hip_runtime.h>
#include <hip/hip_bf16.h>

typedef __attribute__((ext_vector_type(2))) float v2f;
typedef __attribute__((ext_vector_type(8))) float v8f;

#define DIM 128
#define EPS 1e-5f

// ---------------------------------------------------------------- zero
__global__ void k_zero(float* __restrict__ p, int n) {
    int g = blockIdx.x * blockDim.x + threadIdx.x;
    if (g < n) p[g] = 0.0f;
}

// ---------------------------------------------------------------- scatter-add aggregation
// agg[dst[e]] += h[src[e]]   (E edges, DIM features)
// thread = (edge, 16-byte feature chunk)
__global__ void k_scatter(const float* __restrict__ h,
                          const int* __restrict__ src,
                          const int* __restrict__ dst,
                          float* __restrict__ agg, int E_) {
    long long gid = (long long)blockIdx.x * blockDim.x + threadIdx.x;
    long long total = (long long)E_ * 32;
    if (gid >= total) return;
    int e  = (int)(gid >> 5);
    int f  = (int)(gid & 31) * 4;
    int s  = src[e];
    int d  = dst[e];
    float4 v = *(const float4*)(h + (size_t)s * DIM + f);
    float* base = agg + (size_t)d * DIM + f;
    atomicAdd(base + 0, v.x);
    atomicAdd(base + 1, v.y);
    atomicAdd(base + 2, v.z);
    atomicAdd(base + 3, v.w);
}

// ---------------------------------------------------------------- fused dual GEMM + bias + BN stats
// hout = agg @ Wr + h @ Ws + bias ; also accumulate per-column sum / sumsq.
// 256 threads = 8 waves; wave w owns output columns [16w, 16w+16).
// B fragments (both weight matrices, full K=128) are held in registers per wave.
// A tiles (16 rows x 128 K, for agg and h) staged in LDS, loaded as b64 fragments.
// Two independent accumulator chains (agg@Wr and h@Ws) for 2-way WMMA ILP.
__global__ __launch_bounds__(256) void k_gemm_dual(
    const float* __restrict__ agg, const float* __restrict__ h,
    const float* __restrict__ Wr,  const float* __restrict__ Ws,
    const float* __restrict__ bias, float* __restrict__ hout,
    float* __restrict__ gsum, float* __restrict__ gsq, int nRowBlocks) {

    __shared__ float sA[16 * DIM];
    __shared__ float sH[16 * DIM];
    __shared__ float ssum[DIM];
    __shared__ float ssq[DIM];

    const int tid   = threadIdx.x;
    const int wave  = tid >> 5;        // 0..7
    const int lane  = tid & 31;
    const int nsub  = lane & 15;       // column within 16-wide tile
    const int khalf = lane >> 4;       // 0: K{0,1}, 1: K{2,3} of each 4-step
    const int col   = wave * 16 + nsub;

    // Preload B fragments for this wave's 16 columns (both weight matrices).
    // B(4x16) layout: vgpr0 = rows K{0,2}, vgpr1 = rows K{1,3}; lane -> (khalf, nsub).
    v2f br[32], bs[32];
#pragma unroll
    for (int kk = 0; kk < 32; ++kk) {
        int krow = kk * 4 + khalf * 2;
        v2f t;
        t.x = Wr[(size_t)krow * DIM + col];
        t.y = Wr[(size_t)(krow + 1) * DIM + col];
        br[kk] = t;
        t.x = Ws[(size_t)krow * DIM + col];
        t.y = Ws[(size_t)(krow + 1) * DIM + col];
        bs[kk] = t;
    }
    const float bcol = bias[col];

    if (tid < DIM) { ssum[tid] = 0.0f; ssq[tid] = 0.0f; }

    for (int rb = blockIdx.x; rb < nRowBlocks; rb += gridDim.x) {
        __syncthreads();  // protect LDS reuse across iterations
        const int row0 = rb * 16;
        {   // coalesced stage of 16x128 tiles of agg and h (2048 floats each)
            int base = tid * 8;
            int r = base >> 7;
            int c = base & (DIM - 1);
            const float4* ga = (const float4*)(agg + (size_t)(row0 + r) * DIM + c);
            const float4* gh = (const float4*)(h   + (size_t)(row0 + r) * DIM + c);
            float4* la = (float4*)(sA + base);
            float4* lh = (float4*)(sH + base);
            la[0] = ga[0]; la[1] = ga[1];
            lh[0] = gh[0]; lh[1] = gh[1];
        }
        __syncthreads();

        v8f accR = {0.f, 0.f, 0.f, 0.f, 0.f, 0.f, 0.f, 0.f};   // agg @ Wr chain
        v8f accS = {0.f, 0.f, 0.f, 0.f, 0.f, 0.f, 0.f, 0.f};   // h   @ Ws chain
        // A(16x4) fragment for lane: row M=nsub, K pair at (4kk + 2*khalf)
        const int aoff = nsub * DIM + khalf * 2;
#pragma unroll
        for (int kk = 0; kk < 32; ++kk) {
            v2f a1 = *(const v2f*)(sA + aoff + kk * 4);
            accR = __builtin_amdgcn_wmma_f32_16x16x4_f32(
                false, a1, false, br[kk], (short)0, accR, false, false);
            v2f a2 = *(const v2f*)(sH + aoff + kk * 4);
            accS = __builtin_amdgcn_wmma_f32_16x16x4_f32(
                false, a2, false, bs[kk], (short)0, accS, false, false);
        }

        // epilogue: combine chains, bias, store, column stats
        float lsum = 0.f, lsq = 0.f;
#pragma unroll
        for (int i = 0; i < 8; ++i) {
            float v = accR[i] + accS[i] + bcol;
            // C/D layout: vgpr i -> row M = i + 8*khalf, col N = nsub
            hout[(size_t)(row0 + i + 8 * khalf) * DIM + col] = v;
            lsum += v;
            lsq  += v * v;
        }
        lsum += __shfl_xor(lsum, 16, 32);
        lsq  += __shfl_xor(lsq, 16, 32);
        if (khalf == 0) {       // one lane per column within the wave
            ssum[col] += lsum;
            ssq[col]  += lsq;
        }
    }
    __syncthreads();
    if (tid < DIM) {
        atomicAdd(&gsum[tid], ssum[tid]);
        atomicAdd(&gsq[tid],  ssq[tid]);
    }
}

// ---------------------------------------------------------------- BN finalize (biased variance)
__global__ void k_bn_finalize(const float* __restrict__ gsum, const float* __restrict__ gsq,
                              const float* __restrict__ gamma, const float* __restrict__ beta,
                              float* __restrict__ scale, float* __restrict__ shift, float invN) {
    int c = threadIdx.x;
    float mu  = gsum[c] * invN;
    float var = gsq[c] * invN - mu * mu;
    float sc  = gamma[c] * rsqrtf(var + EPS);
    scale[c] = sc;
    shift[c] = beta[c] - mu * sc;
}

// ---------------------------------------------------------------- affine + relu (float4)
__global__ void k_bn_relu(float* __restrict__ h, const float* __restrict__ scale,
                          const float* __restrict__ shift, int nQuads) {
    int g = blockIdx.x * blockDim.x + threadIdx.x;
    if (g >= nQuads) return;
    int c = (g * 4) & (DIM - 1);
    float4 v = *(float4*)(h + (size_t)g * 4);
    v.x = fmaxf(v.x * scale[c + 0] + shift[c + 0], 0.f);
    v.y = fmaxf(v.y * scale[c + 1] + shift[c + 1], 0.f);
    v.z = fmaxf(v.z * scale[c + 2] + shift[c + 2], 0.f);
    v.w = fmaxf(v.w * scale[c + 3] + shift[c + 3], 0.f);
    *(float4*)(h + (size_t)g * 4) = v;
}

// ---------------------------------------------------------------- final layer: out = agg@W1 + h@W2 + b2  (OUT=2)
__global__ __launch_bounds__(256) void k_final(const float* __restrict__ agg,
                                               const float* __restrict__ h,
                                               const float* __restrict__ W1,
                                               const float* __restrict__ W2,
                                               const float* __restrict__ b2,
                                               float* __restrict__ out, int n) {
    int node = blockIdx.x * 8 + (threadIdx.x >> 5);
    int lane = threadIdx.x & 31;
    if (node >= n) return;
    float a0 = 0.f, a1 = 0.f;
#pragma unroll
    for (int k = lane; k < DIM; k += 32) {
        float av = agg[(size_t)node * DIM + k];
        float hv = h[(size_t)node * DIM + k];
        a0 += av * W1[k * 2 + 0] + hv * W2[k * 2 + 0];
        a1 += av * W1[k * 2 + 1] + hv * W2[k * 2 + 1];
    }
#pragma unroll
    for (int off = 16; off > 0; off >>= 1) {
        a0 += __shfl_down(a0, off, 32);
        a1 += __shfl_down(a1, off, 32);
    }
    if (lane == 0) {
        out[(size_t)node * 2 + 0] = a0 + b2[0];
        out[(size_t)node * 2 + 1] = a1 + b2[1];
    }
}

// ================================================================ launch
extern "C" void kernel_launch(void* const* d_in, const int* in_sizes, int n_in,
                              void* d_out, int out_size, void* d_ws, size_t ws_size,
                              hipStream_t stream) {
    const float* x     = (const float*)d_in[0];
    const int*   ei    = (const int*)d_in[1];
    const float* Wrel  = (const float*)d_in[2];
    const float* Wroot = (const float*)d_in[3];
    const float* b     = (const float*)d_in[4];
    const float* gamma = (const float*)d_in[5];
    const float* beta  = (const float*)d_in[6];
    const float* Wrel2 = (const float*)d_in[7];
    const float* Wroot2= (const float*)d_in[8];
    const float* b2    = (const float*)d_in[9];
    float* out = (float*)d_out;

    const int N_ = in_sizes[0] / DIM;
    const int E_ = in_sizes[1] / 2;
    const int* src = ei;
    const int* dst = ei + E_;

    const size_t ND = (size_t)N_ * DIM;
    float* hA    = (float*)d_ws;
    float* hB    = hA + ND;
    float* agg   = hB + ND;
    float* stats = agg + ND;          // [0:128)=sum [128:256)=sq [256:384)=scale [384:512)=shift
    float* gsum  = stats;
    float* gsq   = stats + DIM;
    float* scale = stats + 2 * DIM;
    float* shift = stats + 3 * DIM;

    const int nRB = N_ / 16;
    const int zeroN = (int)(ND + 4 * DIM);            // agg + stats are contiguous
    const int zeroBlocks = (zeroN + 255) / 256;
    const long long scTotal = (long long)E_ * 32;
    const int scBlocks = (int)((scTotal + 255) / 256);
    const int gemmBlocks = (nRB < 624) ? nRB : 624;
    const int nQuads = (int)(ND / 4);
    const int brBlocks = (nQuads + 255) / 256;
    const float invN = 1.0f / (float)N_;

    const float* hin = x;
    float* hping[3] = {hA, hB, hA};
    for (int li = 0; li < 3; ++li) {
        float* hout = hping[li];
        k_zero<<<zeroBlocks, 256, 0, stream>>>(agg, zeroN);
        k_scatter<<<scBlocks, 256, 0, stream>>>(hin, src, dst, agg, E_);
        k_gemm_dual<<<gemmBlocks, 256, 0, stream>>>(
            agg, hin, Wrel + (size_t)li * DIM * DIM, Wroot + (size_t)li * DIM * DIM,
            b + (size_t)li * DIM, hout, gsum, gsq, nRB);
        k_bn_finalize<<<1, DIM, 0, stream>>>(gsum, gsq, gamma + (size_t)li * DIM,
                                             beta + (size_t)li * DIM, scale, shift, invN);
        k_bn_relu<<<brBlocks, 256, 0, stream>>>(hout, scale, shift, nQuads);
        hin = hout;
    }

    // final GraphConv to OUT=2
    k_zero<<<zeroBlocks, 256, 0, stream>>>(agg, (int)ND);
    k_scatter<<<scBlocks, 256, 0, stream>>>(hin, src, dst, agg, E_);
    k_final<<<(N_ + 7) / 8, 256, 0, stream>>>(agg, hin, Wrel2, Wroot2, b2, out, N_);
}